// GCN2Net_35167192220486
// MI455X (gfx1250) — compile-verified
//
#include <hip/hip_runtime.h>
#include <cmath>

typedef float v2f __attribute__((ext_vector_type(2)));
typedef float v8f __attribute__((ext_vector_type(8)));

#define HID 128

// ---------------------------------------------------------------- CSR build
__global__ void k_init_cnt(int* __restrict__ cnt, int N) {
    int i = blockIdx.x * 256 + threadIdx.x;
    if (i < N) cnt[i] = 1;   // self-loop
}

__global__ void k_count(const long long* __restrict__ ei, int* __restrict__ cnt, int E) {
    int e = blockIdx.x * 256 + threadIdx.x;
    if (e >= E) return;
    int c = (int)ei[(size_t)E + e];          // col = edge_index[1][e]
    atomicAdd(&cnt[c], 1);
}

// single-block chunked inclusive scan -> exclusive offsets[0..N]
__global__ __launch_bounds__(1024) void k_scan(const int* __restrict__ cnt,
                                               int* __restrict__ offs, int N) {
    __shared__ int sh[1024];
    int tid = threadIdx.x;
    int carry = 0;
    if (tid == 0) offs[0] = 0;
    for (int base = 0; base < N; base += 1024) {
        int i = base + tid;
        sh[tid] = (i < N) ? cnt[i] : 0;
        __syncthreads();
        for (int off = 1; off < 1024; off <<= 1) {
            int t = (tid >= off) ? sh[tid - off] : 0;
            __syncthreads();
            sh[tid] += t;
            __syncthreads();
        }
        if (i < N) offs[i + 1] = carry + sh[tid];
        carry += sh[1023];
        __syncthreads();
    }
}

__global__ void k_prep_nodes(const int* __restrict__ cnt, const int* __restrict__ offs,
                             float* __restrict__ dinv, int* __restrict__ cursor,
                             int* __restrict__ csr_src, float* __restrict__ csr_w, int N) {
    int i = blockIdx.x * 256 + threadIdx.x;
    if (i >= N) return;
    float di = rsqrtf((float)cnt[i]);   // cnt >= 1 always
    dinv[i] = di;
    int o = offs[i];
    csr_src[o] = i;                     // self-loop entry
    csr_w[o]   = di;
    cursor[i] = o + 1;
}

__global__ void k_fill(const long long* __restrict__ ei, const float* __restrict__ dinv,
                       int* __restrict__ cursor, int* __restrict__ csr_src,
                       float* __restrict__ csr_w, int E) {
    int e = blockIdx.x * 256 + threadIdx.x;
    if (e >= E) return;
    int r = (int)ei[e];
    int c = (int)ei[(size_t)E + e];
    int pos = atomicAdd(&cursor[c], 1);
    csr_src[pos] = r;
    csr_w[pos]   = dinv[r];
}

// ---------------------------------------------------------------- propagation
// one wave (32 lanes) per node; lane owns 4 consecutive features (float4).
// s[v] = (1-alpha) * dinv[v] * sum_r dinv[r]*h[r]   +   alpha * h0[v]
// Node id forced to SGPR via readfirstlane -> per-edge index/weight fetches
// become scalar (SMEM) loads; h-row gathers get an SGPR base.
__global__ __launch_bounds__(256) void k_gather(const float* __restrict__ h,
                                                const float* __restrict__ h0,
                                                const float* __restrict__ dinv,
                                                const int* __restrict__ offs,
                                                const int* __restrict__ csr_src,
                                                const float* __restrict__ csr_w,
                                                float* __restrict__ s_out, int N) {
    int gid  = blockIdx.x * 256 + threadIdx.x;
    int v    = __builtin_amdgcn_readfirstlane(gid >> 5);   // wave-uniform node id
    int lane = threadIdx.x & 31;
    if (v >= N) return;
    int beg = offs[v], end = offs[v + 1];
    float ax = 0.f, ay = 0.f, az = 0.f, aw = 0.f;
    int e = beg;
    for (; e + 1 < end; e += 2) {
        int   r0 = csr_src[e],  r1 = csr_src[e + 1];
        float w0 = csr_w[e],    w1 = csr_w[e + 1];
        float4 p = ((const float4*)(h + (size_t)r0 * HID))[lane];
        float4 q = ((const float4*)(h + (size_t)r1 * HID))[lane];
        ax += w0 * p.x + w1 * q.x;  ay += w0 * p.y + w1 * q.y;
        az += w0 * p.z + w1 * q.z;  aw += w0 * p.w + w1 * q.w;
    }
    if (e < end) {
        int   r0 = csr_src[e];
        float w0 = csr_w[e];
        float4 p = ((const float4*)(h + (size_t)r0 * HID))[lane];
        ax += w0 * p.x; ay += w0 * p.y; az += w0 * p.z; aw += w0 * p.w;
    }
    float scale = 0.9f * dinv[v];   // (1 - ALPHA) * dinv[v]
    float4 r0v = ((const float4*)(h0 + (size_t)v * HID))[lane];
    float4 o;
    o.x = scale * ax + 0.1f * r0v.x;
    o.y = scale * ay + 0.1f * r0v.y;
    o.z = scale * az + 0.1f * r0v.z;
    o.w = scale * aw + 0.1f * r0v.w;
    ((float4*)(s_out + (size_t)v * HID))[lane] = o;
}

// ---------------------------------------------------------------- WMMA GEMM
// Out = post( a_res * R + b_mm * (A @ W) + bias ), optional relu, optional Out2 copy.
// Block: 256 threads = 8 waves; wave w handles row-tile (blockIdx*8 + w) = 16 rows x 128 cols.
// W staged in LDS pre-swizzled into K-pairs: Ws2[k2][n] = {W[2*k2][n], W[2*k2+1][n]}
// so each B fragment is a single aligned ds_load_b64 into an even VGPR pair.
__global__ __launch_bounds__(256) void k_gemm(const float* __restrict__ A,
                                              const float* __restrict__ W,
                                              const float* __restrict__ bias,
                                              const float* __restrict__ R,
                                              float* __restrict__ Out,
                                              float* __restrict__ Out2,
                                              float a_res, float b_mm, int M, int do_relu) {
    __shared__ float Ws2[HID * HID];   // [k2=64][n=128] of float2, flat floats = 64KB
    for (int t = threadIdx.x; t < HID * HID / 4; t += 256) {
        float4 w4 = ((const float4*)W)[t];
        int lin = t * 4;                  // element (k, n0..n0+3), same k
        int k = lin >> 7;                 // / 128
        int n0 = lin & 127;
        int base2 = ((k >> 1) * HID) * 2 + (k & 1);   // float index of component
        Ws2[base2 + (n0 + 0) * 2] = w4.x;
        Ws2[base2 + (n0 + 1) * 2] = w4.y;
        Ws2[base2 + (n0 + 2) * 2] = w4.z;
        Ws2[base2 + (n0 + 3) * 2] = w4.w;
    }
    __syncthreads();

    int wave = threadIdx.x >> 5;
    int lane = threadIdx.x & 31;
    int rowtile = blockIdx.x * 8 + wave;
    int mtiles = (M + 15) >> 4;
    if (rowtile >= mtiles) return;

    int lm    = lane & 15;          // M / N index within tile
    int khalf = lane >> 4;          // selects K-pair (ISA 16x4 f32 A layout)
    int m0    = rowtile * 16 + lm;

    v8f acc[8];
    #pragma unroll
    for (int i = 0; i < 8; ++i)
        #pragma unroll
        for (int j = 0; j < 8; ++j) acc[i][j] = 0.0f;

    const v2f* Wp = (const v2f*)Ws2;
    #pragma unroll 4
    for (int kt = 0; kt < HID / 4; ++kt) {
        int k2 = kt * 2 + khalf;                       // K-pair index
        v2f afrag = *(const v2f*)(A + (size_t)m0 * HID + k2 * 2);
        #pragma unroll
        for (int nt = 0; nt < 8; ++nt) {
            int n = nt * 16 + lm;
            v2f bfrag = Wp[k2 * HID + n];              // single ds_load_b64
            acc[nt] = __builtin_amdgcn_wmma_f32_16x16x4_f32(
                false, afrag, false, bfrag, (short)0, acc[nt], false, false);
        }
    }

    // C/D layout: element (m = v + 8*khalf, n = lm + 16*nt) in acc[nt][v]
    #pragma unroll
    for (int nt = 0; nt < 8; ++nt) {
        #pragma unroll
        for (int v = 0; v < 8; ++v) {
            int m = rowtile * 16 + v + 8 * khalf;
            if (m >= M) continue;
            int n = nt * 16 + lm;
            float val = b_mm * acc[nt][v];
            if (R)    val += a_res * R[(size_t)m * HID + n];
            if (bias) val += bias[n];
            if (do_relu) val = fmaxf(val, 0.0f);
            Out[(size_t)m * HID + n] = val;
            if (Out2) Out2[(size_t)m * HID + n] = val;
        }
    }
}

// ---------------------------------------------------------------- output head
__global__ __launch_bounds__(256) void k_out(const float* __restrict__ h,
                                             const float* __restrict__ Wout,
                                             const float* __restrict__ bout,
                                             float* __restrict__ out, int N) {
    int gid  = blockIdx.x * 256 + threadIdx.x;
    int v    = gid >> 5;
    int lane = threadIdx.x & 31;
    if (v >= N) return;
    const float* hv = h + (size_t)v * HID;
    float s = 0.f;
    #pragma unroll
    for (int j = 0; j < 4; ++j) s += hv[lane + 32 * j] * Wout[lane + 32 * j];
    #pragma unroll
    for (int off = 16; off > 0; off >>= 1) s += __shfl_down(s, off, 32);
    if (lane == 0) out[v] = s + bout[0];
}

// ---------------------------------------------------------------- launcher
extern "C" void kernel_launch(void* const* d_in, const int* in_sizes, int n_in,
                              void* d_out, int out_size, void* d_ws, size_t ws_size,
                              hipStream_t stream) {
    const float*     x     = (const float*)d_in[0];
    const float*     W_in  = (const float*)d_in[2];
    const float*     b_in  = (const float*)d_in[3];
    const float*     W_gcn = (const float*)d_in[4];
    const float*     W_out = (const float*)d_in[5];
    const float*     b_out = (const float*)d_in[6];
    const long long* ei    = (const long long*)d_in[7];

    const int N      = in_sizes[0] / HID;
    const int E      = in_sizes[7] / 2;
    const int LAYERS = in_sizes[4] / (HID * HID);

    // workspace carve-up (256B aligned)
    char* base = (char*)d_ws;
    size_t off = 0;
    auto carve = [&](size_t bytes) -> char* {
        char* p = base + off;
        off = (off + bytes + 255) & ~(size_t)255;
        return p;
    };
    int*   cnt    = (int*)  carve((size_t)N * 4);
    int*   offs   = (int*)  carve((size_t)(N + 1) * 4);
    int*   cursor = (int*)  carve((size_t)N * 4);
    float* dinv   = (float*)carve((size_t)N * 4);
    int*   csr    = (int*)  carve((size_t)(E + N) * 4);
    float* csrw   = (float*)carve((size_t)(E + N) * 4);
    float* h      = (float*)carve((size_t)N * HID * 4);
    float* h0     = (float*)carve((size_t)N * HID * 4);
    float* s      = (float*)carve((size_t)N * HID * 4);

    const int gN   = (N + 255) / 256;
    const int gE   = (E + 255) / 256;
    const int gWav = (N + 7) / 8;                       // one wave per node
    const int gMM  = ((N + 15) / 16 + 7) / 8;           // 8 row-tiles per block

    // --- degree + CSR build (per call; deterministic work, atomics only affect order)
    k_init_cnt<<<gN, 256, 0, stream>>>(cnt, N);
    k_count   <<<gE, 256, 0, stream>>>(ei, cnt, E);
    k_scan    <<<1, 1024, 0, stream>>>(cnt, offs, N);
    k_prep_nodes<<<gN, 256, 0, stream>>>(cnt, offs, dinv, cursor, csr, csrw, N);
    k_fill    <<<gE, 256, 0, stream>>>(ei, dinv, cursor, csr, csrw, E);

    // --- initial projection: h = h0 = x @ W_in + b_in
    k_gemm<<<gMM, 256, 0, stream>>>(x, W_in, b_in, nullptr, h, h0,
                                    0.0f, 1.0f, N, 0);

    // --- GCNII layers
    for (int i = 0; i < LAYERS; ++i) {
        k_gather<<<gWav, 256, 0, stream>>>(h, h0, dinv, offs, csr, csrw, s, N);
        float beta = logf(0.5f / (float)(i + 1) + 1.0f);
        k_gemm<<<gMM, 256, 0, stream>>>(s, W_gcn + (size_t)i * HID * HID, nullptr, s,
                                        h, nullptr, 1.0f - beta, beta, N, 1);
    }

    // --- output head
    k_out<<<gWav, 256, 0, stream>>>(h, W_out, b_out, (float*)d_out, N);
}